// GHOPContactRefinement_87385404604975
// MI455X (gfx1250) — compile-verified
//
#include <hip/hip_runtime.h>
#include <hip/hip_bf16.h>
#include <math.h>

typedef __attribute__((ext_vector_type(2))) float v2f;
typedef __attribute__((ext_vector_type(8))) float v8f;

#define BATCH 8
#define NH 778
#define NTILE_N ((NH + 15) / 16)   // 49 row tiles
#define CTHRESH 0.01f
#define PTHRESH 0.005f

// ---------------------------------------------------------------------------
// Kernel 1: build augmented B' = [-2x, -2y, -2z, |b|^2] per object vertex.
// Padded rows (n >= V) get |b|^2 = 3e38 so they can never win the min.
// ---------------------------------------------------------------------------
__global__ __launch_bounds__(256)
void ghop_prep_bprime(const float* __restrict__ obj, float* __restrict__ bp,
                      int V, int Mpad) {
    int i = blockIdx.x * blockDim.x + threadIdx.x;   // over BATCH * Mpad
    if (i >= BATCH * Mpad) return;
    int b = i / Mpad, m = i % Mpad;
    float4 o;
    if (m < V) {
        const float* p = obj + ((size_t)b * V + m) * 3;
        float x = p[0], y = p[1], z = p[2];
        o = make_float4(-2.0f * x, -2.0f * y, -2.0f * z, x * x + y * y + z * z);
    } else {
        o = make_float4(0.0f, 0.0f, 0.0f, 3.0e38f);
    }
    ((float4*)bp)[(size_t)b * Mpad + m] = o;
}

// ---------------------------------------------------------------------------
// Kernel 2: WMMA 1-NN. One block = one (batch, 16-hand-row tile).
// 8 waves per block stride over the object tiles (wave-uniform t -> EXEC all 1s).
//   A' (16x4, f32, 2 VGPR): lanes 0-15 hold K=0,1 (x,y); lanes 16-31 K=2,3 (z,1)
//   B' (4x16, f32, 2 VGPR): lanes 0-15 hold K=0,1; lanes 16-31 K=2,3
//   D  (16x16 f32, 8 VGPR): VGPR i -> rows i (lanes 0-15) and i+8 (lanes 16-31)
// d^2 = |a|^2 + min_n (A' x B')  ; |a|^2 added after the min.
// Depth-1 software pipeline + raw v_min_num_f32 accumulation (no canonicalize).
// ---------------------------------------------------------------------------
__global__ __launch_bounds__(256)
void ghop_nn_wmma(const float* __restrict__ hv, const float* __restrict__ bp,
                  float* __restrict__ dists, int Mpad) {
    __shared__ float smin[8][16];

    const int lane = threadIdx.x & 31;
    const int wave = threadIdx.x >> 5;
    const int tile = blockIdx.x % NTILE_N;
    const int b    = blockIdx.x / NTILE_N;
    const int row0 = tile * 16;
    const int half = lane >> 4;     // 0: K=0,1   1: K=2,3
    const int l16  = lane & 15;

    // A' tile (identical in every wave of the block)
    v2f a;
    {
        int row = row0 + l16;
        if (row < NH) {
            const float* p = hv + ((size_t)b * NH + row) * 3;
            if (half == 0) { a.x = p[0]; a.y = p[1]; }
            else           { a.x = p[2]; a.y = 1.0f; }
        } else {
            a.x = 0.0f; a.y = 0.0f;   // padded row -> dot = 0, discarded later
        }
    }

    const int mtiles = Mpad >> 4;                              // 2500
    const float2* bpt = (const float2*)bp + (size_t)b * Mpad * 2;
    const size_t laneoff = (size_t)l16 * 2 + half;

    float rm[8];
#pragma unroll
    for (int i = 0; i < 8; ++i) rm[i] = 3.0e38f;

    // ---- software-pipelined main loop: prefetch next B' tile before use ----
    int t = wave;                                   // wave-uniform
    float2 bv = bpt[((size_t)t << 5) + laneoff];    // tile t, this lane's K-pair
    while (t + 8 < mtiles) {
        float2 nx = bpt[((size_t)(t + 8) << 5) + laneoff];
        v2f bb; bb.x = bv.x; bb.y = bv.y;
        v8f c = {};
        v8f d = __builtin_amdgcn_wmma_f32_16x16x4_f32(
            false, a, false, bb, (short)0, c, false, false);
#pragma unroll
        for (int i = 0; i < 8; ++i)
            asm volatile("v_min_num_f32 %0, %0, %1" : "+v"(rm[i]) : "v"(d[i]));
        bv = nx;
        t += 8;
    }
    // ---- epilogue: last tile ----
    {
        v2f bb; bb.x = bv.x; bb.y = bv.y;
        v8f c = {};
        v8f d = __builtin_amdgcn_wmma_f32_16x16x4_f32(
            false, a, false, bb, (short)0, c, false, false);
#pragma unroll
        for (int i = 0; i < 8; ++i)
            asm volatile("v_min_num_f32 %0, %0, %1" : "+v"(rm[i]) : "v"(d[i]));
    }

    // min across the 16 columns: xor masks 1,2,4,8 stay inside each 16-lane half
#pragma unroll
    for (int m = 1; m <= 8; m <<= 1) {
#pragma unroll
        for (int i = 0; i < 8; ++i) {
            float o = __shfl_xor(rm[i], m, 32);
            rm[i] = fminf(rm[i], o);
        }
    }
    if (l16 == 0) {
#pragma unroll
        for (int i = 0; i < 8; ++i) smin[wave][i + half * 8] = rm[i];
    }
    __syncthreads();

    if (threadIdx.x < 16) {
        int r = threadIdx.x;
        float mn = smin[0][r];
#pragma unroll
        for (int w = 1; w < 8; ++w) mn = fminf(mn, smin[w][r]);
        int row = row0 + r;
        if (row < NH) {
            const float* p = hv + ((size_t)b * NH + row) * 3;
            float asq = p[0] * p[0] + p[1] * p[1] + p[2] * p[2];
            dists[(size_t)b * NH + row] = sqrtf(fmaxf(mn + asq, 0.0f));
        }
    }
}

// ---------------------------------------------------------------------------
// Kernel 3: masked-mean reductions -> 6 scalars.
// contact_dists[b,j] == dists[b, CONTACT_IDX[j]] (row subset of same NN).
// ---------------------------------------------------------------------------
__global__ __launch_bounds__(256)
void ghop_reduce_loss(const float* __restrict__ dists, float* __restrict__ out) {
    const int cidx[10] = {745, 317, 444, 556, 673, 95, 182, 234, 279, 320};
    __shared__ float s_pen[256], s_all[256], s_att[256];
    __shared__ int   s_pcnt[256], s_acnt[256];
    int tid = threadIdx.x;
    float pen = 0.0f, alls = 0.0f, att = 0.0f;
    int pcnt = 0, acnt = 0;

    for (int i = tid; i < BATCH * NH; i += 256) {
        float d = dists[i];
        alls += d;
        if (d < PTHRESH) { float e = PTHRESH - d; pen += e * e; pcnt++; }
    }
    if (tid < BATCH * 10) {
        int b = tid / 10, j = tid % 10;
        float d = dists[b * NH + cidx[j]];
        if (d > PTHRESH && d < CTHRESH) { att += d * d; acnt++; }
    }
    s_pen[tid] = pen; s_all[tid] = alls; s_att[tid] = att;
    s_pcnt[tid] = pcnt; s_acnt[tid] = acnt;
    __syncthreads();
    for (int s = 128; s > 0; s >>= 1) {
        if (tid < s) {
            s_pen[tid]  += s_pen[tid + s];
            s_all[tid]  += s_all[tid + s];
            s_att[tid]  += s_att[tid + s];
            s_pcnt[tid] += s_pcnt[tid + s];
            s_acnt[tid] += s_acnt[tid + s];
        }
        __syncthreads();
    }
    if (tid == 0) {
        float pen_loss = s_pcnt[0] > 0 ? s_pen[0] / (float)s_pcnt[0] : 0.0f;
        float att_loss = s_acnt[0] > 0 ? s_att[0] / (float)s_acnt[0] : 0.0f;
        out[0] = 100.0f * pen_loss + 10.0f * att_loss;  // contact_loss
        out[1] = pen_loss;
        out[2] = att_loss;
        out[3] = s_all[0] / (float)(BATCH * NH);        // dist_mean
        out[4] = (float)s_acnt[0];                      // num_contacts
        out[5] = (float)s_pcnt[0];                      // num_penetrations
    }
}

extern "C" void kernel_launch(void* const* d_in, const int* in_sizes, int n_in,
                              void* d_out, int out_size, void* d_ws, size_t ws_size,
                              hipStream_t stream) {
    const float* hv = (const float*)d_in[0];          // [8,778,3]
    const float* ov = (const float*)d_in[1];          // [8,V,3]
    int V    = in_sizes[1] / (BATCH * 3);             // 40000
    int Mpad = (V + 15) & ~15;

    float* bp    = (float*)d_ws;                      // B' : BATCH*Mpad*4 floats
    float* dists = bp + (size_t)BATCH * Mpad * 4;     // dists: BATCH*NH floats

    int n1 = BATCH * Mpad;
    ghop_prep_bprime<<<(n1 + 255) / 256, 256, 0, stream>>>(ov, bp, V, Mpad);
    ghop_nn_wmma<<<BATCH * NTILE_N, 256, 0, stream>>>(hv, bp, dists, Mpad);
    ghop_reduce_loss<<<1, 256, 0, stream>>>(dists, (float*)d_out);
}